// GCN2D_86998857548336
// MI455X (gfx1250) — compile-verified
//
#include <hip/hip_runtime.h>

// GCN forward: 3x (GEMM 128x128 via V_WMMA_F32_16X16X4_F32 + edge gather/scatter
// with f32 atomics + self-loop/bias/ReLU epilogue), then mean-pool + classifier.

#define NN 100000   // nodes
#define EE 1600000  // edges
#define HH 128      // feature width (FIN == H == 128)
#define GG 64       // graphs
#define CC 2        // classes

typedef __attribute__((ext_vector_type(2))) float v2f;
typedef __attribute__((ext_vector_type(8))) float v8f;

// ---------------- degree / normalization ----------------

__global__ __launch_bounds__(256) void init_deg_kernel(float* __restrict__ deg) {
  int i = blockIdx.x * 256 + threadIdx.x;
  if (i < NN) deg[i] = 1.0f;  // self-loop weight
}

__global__ __launch_bounds__(256) void edge_deg_kernel(const int* __restrict__ dst,
                                                       const float* __restrict__ ew,
                                                       float* __restrict__ deg) {
  int e = blockIdx.x * 256 + threadIdx.x;
  if (e < EE) atomicAdd(deg + dst[e], ew[e]);
}

__global__ __launch_bounds__(256) void dinv_kernel(float* __restrict__ deg,
                                                   float* __restrict__ snorm) {
  int i = blockIdx.x * 256 + threadIdx.x;
  if (i >= NN) return;
  float d = deg[i];
  float r = d > 0.0f ? __frsqrt_rn(d) : 0.0f;
  deg[i] = r;            // deg buffer now holds dinv
  snorm[i] = r * r;      // self-loop norm (ew = 1)
}

__global__ __launch_bounds__(256) void edge_norm_kernel(const int* __restrict__ src,
                                                        const int* __restrict__ dst,
                                                        const float* __restrict__ ew,
                                                        const float* __restrict__ dinv,
                                                        float* __restrict__ norm) {
  int e = blockIdx.x * 256 + threadIdx.x;
  if (e < EE) norm[e] = dinv[src[e]] * ew[e] * dinv[dst[e]];
}

// ---------------- GEMM: Y[NN x 128] = X[NN x 128] @ W[128 x 128] ----------------
// Block = 256 threads = 8 waves; each wave owns one 16x16 output tile of a
// 16-row strip. K loop: 32 x V_WMMA_F32_16X16X4_F32.

__global__ __launch_bounds__(256) void wmma_gemm128_kernel(const float* __restrict__ X,
                                                           const float* __restrict__ W,
                                                           float* __restrict__ Y) {
  const int lane  = threadIdx.x & 31;
  const int wave  = threadIdx.x >> 5;
  const int mBase = blockIdx.x << 4;
  const int l16   = lane & 15;
  const int half  = lane >> 4;          // 0 or 1
  const int nCol  = (wave << 4) + l16;  // output column
  const int mRow  = mBase + l16;        // A row for this lane
  const int kHalf = half << 1;          // K offset 0 or 2 within each 4-step

  const float* arow = X + (size_t)mRow * HH + kHalf;          // A[m][k..]
  const float* bcol = W + (size_t)kHalf * HH + nCol;          // B[k..][n]

  v8f acc = {};
#pragma unroll 4
  for (int k = 0; k < HH; k += 4) {
    v2f a, b;
    a.x = arow[k];
    a.y = arow[k + 1];
    b.x = bcol[(size_t)k * HH];
    b.y = bcol[(size_t)(k + 1) * HH];
    acc = __builtin_amdgcn_wmma_f32_16x16x4_f32(
        /*neg_a=*/false, a, /*neg_b=*/false, b,
        /*c_mod=*/(short)0, acc, /*reuse_a=*/false, /*reuse_b=*/false);
  }

  const int rBase = mBase + (half << 3);  // rows rBase..rBase+7
#pragma unroll
  for (int v = 0; v < 8; ++v)
    Y[(size_t)(rBase + v) * HH + nCol] = acc[v];
}

// ---------------- aggregation ----------------

__global__ __launch_bounds__(256) void bias_init_kernel(float* __restrict__ agg,
                                                        const float* __restrict__ b) {
  long long i = (long long)blockIdx.x * 256 + threadIdx.x;
  if (i < (long long)NN * HH) agg[i] = b[i & (HH - 1)];
}

// One wave per edge: gather 512B row, scale by norm, scatter-add with f32 atomics.
__global__ __launch_bounds__(256) void edge_scatter_kernel(const float* __restrict__ xw,
                                                           const int* __restrict__ src,
                                                           const int* __restrict__ dst,
                                                           const float* __restrict__ norm,
                                                           float* __restrict__ agg) {
  int e = blockIdx.x * 8 + (threadIdx.x >> 5);
  if (e >= EE) return;
  const int lane = threadIdx.x & 31;
  const int s = src[e];
  const int d = dst[e];
  const float w = norm[e];
  const float4 m = *((const float4*)(xw + (size_t)s * HH) + lane);
  float* q = agg + (size_t)d * HH + lane * 4;
  atomicAdd(q + 0, m.x * w);
  atomicAdd(q + 1, m.y * w);
  atomicAdd(q + 2, m.z * w);
  atomicAdd(q + 3, m.w * w);
}

// Self-loop contribution + ReLU (runs after scatter; rows exclusive -> no atomics).
__global__ __launch_bounds__(256) void self_relu_kernel(float* __restrict__ agg,
                                                        const float* __restrict__ xw,
                                                        const float* __restrict__ snorm) {
  long long i = (long long)blockIdx.x * 256 + threadIdx.x;
  if (i >= (long long)NN * HH) return;
  float v = agg[i] + xw[i] * snorm[i >> 7];
  agg[i] = v > 0.0f ? v : 0.0f;
}

// ---------------- pooling + classifier ----------------

__global__ __launch_bounds__(256) void zero_pool_kernel(float* __restrict__ pooled,
                                                        float* __restrict__ cnt) {
  int i = blockIdx.x * 256 + threadIdx.x;
  if (i < GG * HH) pooled[i] = 0.0f;
  if (i < GG) cnt[i] = 0.0f;
}

__global__ __launch_bounds__(256) void pool_kernel(const float* __restrict__ h,
                                                   const int* __restrict__ batch,
                                                   float* __restrict__ pooled,
                                                   float* __restrict__ cnt) {
  int node = blockIdx.x * 8 + (threadIdx.x >> 5);
  if (node >= NN) return;
  const int lane = threadIdx.x & 31;
  const int g = batch[node];
  const float4 m = *((const float4*)(h + (size_t)node * HH) + lane);
  float* q = pooled + (size_t)g * HH + lane * 4;
  atomicAdd(q + 0, m.x);
  atomicAdd(q + 1, m.y);
  atomicAdd(q + 2, m.z);
  atomicAdd(q + 3, m.w);
  if (lane == 0) atomicAdd(cnt + g, 1.0f);
}

__global__ __launch_bounds__(128) void final_kernel(const float* __restrict__ pooled,
                                                    const float* __restrict__ cnt,
                                                    const float* __restrict__ Wl,
                                                    const float* __restrict__ bl,
                                                    float* __restrict__ out) {
  int t = threadIdx.x;  // 128 threads == G*C
  int g = t >> 1, c = t & 1;
  const float* p = pooled + (size_t)g * HH;
  float s = 0.0f;
#pragma unroll 8
  for (int k = 0; k < HH; ++k) s += p[k] * Wl[k * CC + c];
  float n = cnt[g];
  n = n > 1.0f ? n : 1.0f;
  out[g * CC + c] = s / n + bl[c];
}

// ---------------- launch ----------------

static inline size_t alignup(size_t v) { return (v + 255) & ~(size_t)255; }

extern "C" void kernel_launch(void* const* d_in, const int* in_sizes, int n_in,
                              void* d_out, int out_size, void* d_ws, size_t ws_size,
                              hipStream_t stream) {
  const float* x    = (const float*)d_in[0];    // [N,128]
  const int*   ei   = (const int*)d_in[1];      // [2,E]
  const float* ew   = (const float*)d_in[2];    // [E]
  const int*   bat  = (const int*)d_in[3];      // [N]
  const float* W1   = (const float*)d_in[4];
  const float* b1   = (const float*)d_in[5];
  const float* W2   = (const float*)d_in[6];
  const float* b2   = (const float*)d_in[7];
  const float* W3   = (const float*)d_in[8];
  const float* b3   = (const float*)d_in[9];
  const float* Wl   = (const float*)d_in[10];
  const float* bl   = (const float*)d_in[11];
  float* out = (float*)d_out;

  const int* src = ei;        // edge_index[0]
  const int* dst = ei + EE;   // edge_index[1]

  char* ws = (char*)d_ws;
  float* norm   = (float*)ws; ws += alignup((size_t)EE * 4);
  float* snorm  = (float*)ws; ws += alignup((size_t)NN * 4);
  float* dinv   = (float*)ws; ws += alignup((size_t)NN * 4);   // deg -> dinv in place
  float* buf0   = (float*)ws; ws += alignup((size_t)NN * HH * 4);
  float* buf1   = (float*)ws; ws += alignup((size_t)NN * HH * 4);
  float* buf2   = (float*)ws; ws += alignup((size_t)NN * HH * 4);
  float* pooled = (float*)ws; ws += alignup((size_t)GG * HH * 4);
  float* cnt    = (float*)ws; ws += alignup((size_t)GG * 4);

  const int nBlk  = (NN + 255) / 256;
  const int eBlk  = (EE + 255) / 256;
  const int nhBlk = (int)(((long long)NN * HH + 255) / 256);
  const int mTile = NN / 16;           // 6250 row strips
  const int eWave = (EE + 7) / 8;      // 1 wave per edge
  const int nWave = (NN + 7) / 8;      // 1 wave per node

  // --- normalization ---
  init_deg_kernel<<<nBlk, 256, 0, stream>>>(dinv);
  edge_deg_kernel<<<eBlk, 256, 0, stream>>>(dst, ew, dinv);
  dinv_kernel<<<nBlk, 256, 0, stream>>>(dinv, snorm);
  edge_norm_kernel<<<eBlk, 256, 0, stream>>>(src, dst, ew, dinv, norm);

  // --- layer 1: x -> buf1 ---
  wmma_gemm128_kernel<<<mTile, 256, 0, stream>>>(x, W1, buf0);
  bias_init_kernel<<<nhBlk, 256, 0, stream>>>(buf1, b1);
  edge_scatter_kernel<<<eWave, 256, 0, stream>>>(buf0, src, dst, norm, buf1);
  self_relu_kernel<<<nhBlk, 256, 0, stream>>>(buf1, buf0, snorm);

  // --- layer 2: buf1 -> buf2 ---
  wmma_gemm128_kernel<<<mTile, 256, 0, stream>>>(buf1, W2, buf0);
  bias_init_kernel<<<nhBlk, 256, 0, stream>>>(buf2, b2);
  edge_scatter_kernel<<<eWave, 256, 0, stream>>>(buf0, src, dst, norm, buf2);
  self_relu_kernel<<<nhBlk, 256, 0, stream>>>(buf2, buf0, snorm);

  // --- layer 3: buf2 -> buf1 ---
  wmma_gemm128_kernel<<<mTile, 256, 0, stream>>>(buf2, W3, buf0);
  bias_init_kernel<<<nhBlk, 256, 0, stream>>>(buf1, b3);
  edge_scatter_kernel<<<eWave, 256, 0, stream>>>(buf0, src, dst, norm, buf1);
  self_relu_kernel<<<nhBlk, 256, 0, stream>>>(buf1, buf0, snorm);

  // --- mean pool + classifier ---
  zero_pool_kernel<<<(GG * HH + 255) / 256, 256, 0, stream>>>(pooled, cnt);
  pool_kernel<<<nWave, 256, 0, stream>>>(buf1, bat, pooled, cnt);
  final_kernel<<<1, 128, 0, stream>>>(pooled, cnt, Wl, bl, out);
}